// EfficientAdaptiveThreshold_53025666236668
// MI455X (gfx1250) — compile-verified
//
#include <hip/hip_runtime.h>
#include <hip/hip_bf16.h>

// ---------------------------------------------------------------------------
// EfficientAdaptiveThreshold on MI455X (gfx1250)
// x: [16, 64, 256, 256] f32.  Bandwidth-bound (~520 MB traffic => ~22us at
// 23.3 TB/s).  Channel-mixing GEMV done with chained V_WMMA_F32_16X16X4_F32.
// ---------------------------------------------------------------------------

#define B_DIM 16
#define C_DIM 64
#define HW 65536           // 256*256
#define NUM_BINS 256

typedef float v2f __attribute__((ext_vector_type(2)));
typedef float v8f __attribute__((ext_vector_type(8)));

// Monotonic float <-> uint mapping for atomicMin/atomicMax on floats.
__device__ __forceinline__ unsigned fenc(float f) {
    unsigned u = __float_as_uint(f);
    return (u & 0x80000000u) ? ~u : (u | 0x80000000u);
}
__device__ __forceinline__ float fdec(unsigned u) {
    unsigned b = (u & 0x80000000u) ? (u & 0x7FFFFFFFu) : ~u;
    return __uint_as_float(b);
}

// ---------------------------------------------------------------------------
// Kernel 0: re-initialize workspace accumulators (harness poisons ws once and
// never re-poisons, so every call must reset its own state).
// ---------------------------------------------------------------------------
__global__ void k_init(unsigned* __restrict__ hist, unsigned* __restrict__ minU,
                       unsigned* __restrict__ maxU, float* __restrict__ sig) {
    int idx = blockIdx.x * 256 + threadIdx.x;
    if (idx < B_DIM * NUM_BINS) {
        hist[idx] = 0u;
    } else if (idx < B_DIM * NUM_BINS + 16) {
        minU[idx - B_DIM * NUM_BINS] = 0xFFFFFFFFu;
    } else if (idx < B_DIM * NUM_BINS + 32) {
        maxU[idx - B_DIM * NUM_BINS - 16] = 0u;
    } else if (idx < B_DIM * NUM_BINS + 48) {
        sig[idx - B_DIM * NUM_BINS - 32] = 0.0f;
    }
}

// ---------------------------------------------------------------------------
// Kernel 1: chan_mean[b,c] = mean over HW.  One block per (b,c) plane.
// float4 loads: 512B per wave per instruction, fully coalesced.
// ---------------------------------------------------------------------------
__global__ void k_chan_mean(const float* __restrict__ x, float* __restrict__ cm) {
    int bc = blockIdx.x;                       // 0..1023
    const float4* p4 = (const float4*)(x + (size_t)bc * HW);
    int tid = threadIdx.x;
    float s = 0.0f;
#pragma unroll 4
    for (int i = 0; i < 64; ++i) {             // 64 float4 = 256 floats/thread
        float4 v = p4[tid + (i << 8)];
        s += (v.x + v.y) + (v.z + v.w);
    }
    for (int o = 16; o > 0; o >>= 1) s += __shfl_down(s, o, 32);
    __shared__ float lds[8];
    int wave = tid >> 5, lane = tid & 31;
    if (lane == 0) lds[wave] = s;
    __syncthreads();
    if (tid == 0) {
        float t = 0.0f;
        for (int wv = 0; wv < 8; ++wv) t += lds[wv];
        cm[bc] = t * (1.0f / (float)HW);
    }
}

// ---------------------------------------------------------------------------
// Kernel 2: xn_mean[b,p] = sum_c x[b,c,p] * (chan_mean[b,c]/64) via chained
// V_WMMA_F32_16X16X4_F32, plus streaming sigmoid-sum and min/max per batch.
//
// WMMA mapping (ISA 7.12.2, f32 A 16x4):
//   lane<16 holds A[row=lane, K=0] (v0) and A[row, K=1] (v1)
//   lane>=16 holds A[row=lane-16, K=2] and K=3
// => per chunk of 4 channels each lane loads 2 dwords; lanes 0-15 form one
//    64B-contiguous pixel segment of channel c, lanes 16-31 of channel c+2.
// B[k,n] = w[k] for all n (columns replicated), so D[m,n] = dot(x_row_m, w)
// for every n; lanes 0/16 extract the 16 pixel results from their 8 D VGPRs.
// ---------------------------------------------------------------------------
__global__ void k_wmma_chanmix(const float* __restrict__ x,
                               const float* __restrict__ cm,
                               float* __restrict__ xnm,
                               unsigned* __restrict__ minU,
                               unsigned* __restrict__ maxU,
                               float* __restrict__ sig) {
    int b = blockIdx.y;
    __shared__ float w[C_DIM];
    int tid = threadIdx.x;
    if (tid < C_DIM) w[tid] = cm[b * C_DIM + tid] * (1.0f / (float)C_DIM);
    __syncthreads();

    int wave = tid >> 5, lane = tid & 31;
    int row = lane & 15, hi = lane >> 4;
    const float* xb = x + (size_t)b * C_DIM * HW;
    float* xo = xnm + (size_t)b * HW;
    int pbase = blockIdx.x * 1024;             // 64 pixel-blocks of 1024

    float sAcc = 0.0f, mnL = 3.4e38f, mxL = -3.4e38f;

    for (int t = wave; t < 64; t += 8) {       // 8 tiles of 16 pixels per wave
        int p0 = pbase + t * 16;
        v8f acc = {0.f, 0.f, 0.f, 0.f, 0.f, 0.f, 0.f, 0.f};
#pragma unroll
        for (int kc = 0; kc < 16; ++kc) {      // 16 chunks x 4 channels = 64
            int c0 = (kc << 2) + (hi << 1);
            v2f a, bm;
            a[0] = xb[((size_t)c0 << 16) + p0 + row];
            a[1] = xb[((size_t)(c0 + 1) << 16) + p0 + row];
            bm[0] = w[c0];
            bm[1] = w[c0 + 1];
            acc = __builtin_amdgcn_wmma_f32_16x16x4_f32(
                false, a, false, bm, (short)0, acc, false, false);
        }
        // Columns of D are identical; lane 0 holds pixels p0..p0+7 in its 8
        // D VGPRs, lane 16 holds p0+8..p0+15.
        if (row == 0) {
            int p = p0 + (hi << 3);
            float4 lo4 = make_float4(acc[0], acc[1], acc[2], acc[3]);
            float4 hi4 = make_float4(acc[4], acc[5], acc[6], acc[7]);
            *(float4*)(xo + p) = lo4;
            *(float4*)(xo + p + 4) = hi4;
#pragma unroll
            for (int r = 0; r < 8; ++r) {
                float v = acc[r];
                sAcc += 1.0f / (1.0f + __expf(-v));
                mnL = fminf(mnL, v);
                mxL = fmaxf(mxL, v);
            }
        }
    }

    // Block-level reduction of sigmoid sum and min/max, then merge per batch.
    for (int o = 16; o > 0; o >>= 1) {
        sAcc += __shfl_down(sAcc, o, 32);
        mnL = fminf(mnL, __shfl_down(mnL, o, 32));
        mxL = fmaxf(mxL, __shfl_down(mxL, o, 32));
    }
    __shared__ float rs[8], rmn[8], rmx[8];
    if (lane == 0) { rs[wave] = sAcc; rmn[wave] = mnL; rmx[wave] = mxL; }
    __syncthreads();
    if (tid == 0) {
        float S = 0.0f, MN = 3.4e38f, MX = -3.4e38f;
        for (int wv = 0; wv < 8; ++wv) {
            S += rs[wv];
            MN = fminf(MN, rmn[wv]);
            MX = fmaxf(MX, rmx[wv]);
        }
        atomicAdd(&sig[b], S);
        atomicMin(&minU[b], fenc(MN));
        atomicMax(&maxU[b], fenc(MX));
    }
}

// ---------------------------------------------------------------------------
// Kernel 3: per-batch 256-bin histogram of min-max-normalized xn_mean.
// LDS histogram per block, merged to global with one atomic per bin.
// ---------------------------------------------------------------------------
__global__ void k_hist(const float* __restrict__ xnm,
                       const unsigned* __restrict__ minU,
                       const unsigned* __restrict__ maxU,
                       unsigned* __restrict__ hist) {
    int b = blockIdx.y;
    __shared__ unsigned lh[NUM_BINS];
    int tid = threadIdx.x;
    lh[tid] = 0u;
    __syncthreads();

    float mn = fdec(minU[b]);
    float mx = fdec(maxU[b]);
    float rng = mx - mn;
    if (rng == 0.0f) rng = 1.0f;
    float scale = 255.0f / rng;

    const float* src = xnm + (size_t)b * HW + blockIdx.x * 2048;
    for (int i = tid; i < 2048; i += 256) {
        float v = src[i];
        float norm = fminf(fmaxf((v - mn) * scale, 0.0f), 255.0f);
        int bin = (int)norm;                   // bin_w == 1.0, already clamped
        atomicAdd(&lh[bin], 1u);
    }
    __syncthreads();
    atomicAdd(&hist[b * NUM_BINS + tid], lh[tid]);
}

// ---------------------------------------------------------------------------
// Kernel 4: normalized entropy over nonzero bins + sigmoid mean -> output.
// One block per batch; thread t owns bin t.
// ---------------------------------------------------------------------------
__global__ void k_final(const unsigned* __restrict__ hist,
                        const float* __restrict__ sig,
                        float* __restrict__ out) {
    int b = blockIdx.x, tid = threadIdx.x;
    int wave = tid >> 5, lane = tid & 31;
    float cnt = (float)hist[b * NUM_BINS + tid];

    __shared__ float l1[8];
    __shared__ float sTotal;
    float v = cnt;
    for (int o = 16; o > 0; o >>= 1) v += __shfl_down(v, o, 32);
    if (lane == 0) l1[wave] = v;
    __syncthreads();
    if (tid == 0) {
        float t = 0.0f;
        for (int i = 0; i < 8; ++i) t += l1[i];
        sTotal = t;
    }
    __syncthreads();

    float total = sTotal + 1e-8f;
    float p = cnt / total;
    float term = (p > 0.0f) ? p * __logf(p + 1e-8f) : 0.0f;
    float nzc = (p > 0.0f) ? 1.0f : 0.0f;
    for (int o = 16; o > 0; o >>= 1) {
        term += __shfl_down(term, o, 32);
        nzc += __shfl_down(nzc, o, 32);
    }
    __shared__ float l2[8], l3[8];
    if (lane == 0) { l2[wave] = term; l3[wave] = nzc; }
    __syncthreads();
    if (tid == 0) {
        float T = 0.0f, N = 0.0f;
        for (int i = 0; i < 8; ++i) { T += l2[i]; N += l3[i]; }
        float entropy = -T / fmaxf(N, 1.0f);
        out[b] = sig[b] * (1.0f / (float)HW) + entropy * 10.0f;
    }
}

// ---------------------------------------------------------------------------
// Launch
// ---------------------------------------------------------------------------
extern "C" void kernel_launch(void* const* d_in, const int* in_sizes, int n_in,
                              void* d_out, int out_size, void* d_ws, size_t ws_size,
                              hipStream_t stream) {
    const float* x = (const float*)d_in[0];
    float* out = (float*)d_out;

    // Workspace layout (all offsets 256B-aligned as needed):
    //   [0,      4096)  chan_mean   1024 f32
    //   [4096,   4160)  minU        16 u32
    //   [4160,   4224)  maxU        16 u32
    //   [4224,   4288)  sig_sum     16 f32
    //   [4352,  20736)  hist        4096 u32
    //   [20736, 4215040) xn_mean    1M f32 (4 MB)
    char* ws = (char*)d_ws;
    float*    chan_mean = (float*)ws;
    unsigned* minU      = (unsigned*)(ws + 4096);
    unsigned* maxU      = (unsigned*)(ws + 4160);
    float*    sig       = (float*)(ws + 4224);
    unsigned* hist      = (unsigned*)(ws + 4352);
    float*    xnm       = (float*)(ws + 20736);

    k_init<<<dim3(17), dim3(256), 0, stream>>>(hist, minU, maxU, sig);
    k_chan_mean<<<dim3(B_DIM * C_DIM), dim3(256), 0, stream>>>(x, chan_mean);
    k_wmma_chanmix<<<dim3(64, B_DIM), dim3(256), 0, stream>>>(
        x, chan_mean, xnm, minU, maxU, sig);
    k_hist<<<dim3(32, B_DIM), dim3(256), 0, stream>>>(xnm, minU, maxU, hist);
    k_final<<<dim3(B_DIM), dim3(256), 0, stream>>>(hist, sig, out);
}